// TorchWrithe_44332652430019
// MI455X (gfx1250) — compile-verified
//
#include <hip/hip_runtime.h>
#include <math.h>

// Writhe kernel for MI455X (gfx1250).
// F=256 frames, N_ATOMS=200, S=19503 segments. Output (F,S) f32.
// Store-bandwidth-bound (~20 MB out); frame xyz (2400 B) staged in LDS via
// CDNA5 async-to-LDS loads (ASYNCcnt path), reused by all 8 waves in block.

#define N_ATOMS      200
#define FRAME_FLOATS (N_ATOMS * 3)      // 600 floats
#define FRAME_BYTES  (FRAME_FLOATS * 4) // 2400 bytes = 150 x b128

struct F3 { float x, y, z; };

__device__ __forceinline__ F3 f3(float x, float y, float z) { return {x, y, z}; }
__device__ __forceinline__ F3 sub3(F3 a, F3 b) { return {a.x - b.x, a.y - b.y, a.z - b.z}; }
__device__ __forceinline__ float dot3(F3 a, F3 b) { return a.x * b.x + a.y * b.y + a.z * b.z; }
__device__ __forceinline__ F3 cross3(F3 a, F3 b) {
    return { a.y * b.z - a.z * b.y,
             a.z * b.x - a.x * b.z,
             a.x * b.y - a.y * b.x };
}
// Match reference _nnorm: x / sqrt(sum(x^2))
__device__ __forceinline__ F3 nrm3(F3 v) {
    float inv = 1.0f / sqrtf(dot3(v, v));
    return { v.x * inv, v.y * inv, v.z * inv };
}
__device__ __forceinline__ float clip1(float x) {
    return fminf(1.0f, fmaxf(-1.0f, x));
}

__global__ __launch_bounds__(256) void writhe_kernel(
    const float* __restrict__ xyz,   // (F, 200, 3)
    const int*   __restrict__ segs,  // (S, 4) int32: i, i+1, j, j+1
    float*       __restrict__ out,   // (F, S)
    int S)
{
    __shared__ float sh[FRAME_FLOATS];

    const int f   = blockIdx.x;
    const int tid = threadIdx.x;

    // ---- Stage frame f (2400 B) into LDS with CDNA5 async-to-LDS loads ----
    {
        const float* fbase = xyz + (size_t)f * FRAME_FLOATS;   // uniform -> SGPR pair
        unsigned lds_base  = (unsigned)(size_t)(void*)sh;      // low 32 bits = LDS offset
        if (tid < FRAME_BYTES / 16) {                          // 150 lanes issue b128
            unsigned goff  = (unsigned)tid * 16u;
            unsigned laddr = lds_base + goff;
            asm volatile("global_load_async_to_lds_b128 %0, %1, %2"
                         :: "v"(laddr), "v"(goff), "s"(fbase)
                         : "memory");
        }
        asm volatile("s_wait_asynccnt 0" ::: "memory");        // drain ASYNCcnt
        __syncthreads();
    }

    const float inv2pi = 0.15915494309189535f; // 1/(2*pi)
    const int stride = gridDim.y * blockDim.x;

    for (int k = blockIdx.y * blockDim.x + tid; k < S; k += stride) {
        const int4 sg = ((const int4*)segs)[k];  // coalesced 16B load, L2-hot

        F3 a0 = f3(sh[3 * sg.x], sh[3 * sg.x + 1], sh[3 * sg.x + 2]);
        F3 a1 = f3(sh[3 * sg.y], sh[3 * sg.y + 1], sh[3 * sg.y + 2]);
        F3 a2 = f3(sh[3 * sg.z], sh[3 * sg.z + 1], sh[3 * sg.z + 2]);
        F3 a3 = f3(sh[3 * sg.w], sh[3 * sg.w + 1], sh[3 * sg.w + 2]);

        // disp (reshape order): d0=a2-a0, d1=a3-a0, d2=a2-a1, d3=a3-a1
        F3 d0 = nrm3(sub3(a2, a0));
        F3 d1 = nrm3(sub3(a3, a0));
        F3 d2 = nrm3(sub3(a2, a1));
        F3 d3 = nrm3(sub3(a3, a1));

        // crosses with idx_a=[0,1,3,2], idx_b=[1,3,2,0]
        F3 c0 = nrm3(cross3(d0, d1));
        F3 c1 = nrm3(cross3(d1, d3));
        F3 c2 = nrm3(cross3(d3, d2));
        F3 c3 = nrm3(cross3(d2, d0));

        // dots[k] = crosses[k] . crosses[idx_b[k]]  (faithful to reference,
        // including dots[2] = c2.c2)
        float om = asinf(clip1(dot3(c0, c1)))
                 + asinf(clip1(dot3(c1, c3)))
                 + asinf(clip1(dot3(c2, c2)))
                 + asinf(clip1(dot3(c3, c0)));

        // sign( cross(a3-a2, a1-a0) . d0 )
        float t   = dot3(cross3(sub3(a3, a2), sub3(a1, a0)), d0);
        float sgn = (t > 0.0f) ? 1.0f : ((t < 0.0f) ? -1.0f : 0.0f);

        out[(size_t)f * S + k] = inv2pi * om * sgn;
    }
}

extern "C" void kernel_launch(void* const* d_in, const int* in_sizes, int n_in,
                              void* d_out, int out_size, void* d_ws, size_t ws_size,
                              hipStream_t stream) {
    (void)n_in; (void)out_size; (void)d_ws; (void)ws_size;
    const float* xyz  = (const float*)d_in[0];
    const int*   segs = (const int*)d_in[1];
    float*       out  = (float*)d_out;

    const int S = in_sizes[1] / 4;            // 19503
    const int F = in_sizes[0] / FRAME_FLOATS; // 256

    dim3 grid((unsigned)F, 8, 1);             // 2048 blocks x 8 wave32
    writhe_kernel<<<grid, 256, 0, stream>>>(xyz, segs, out, S);
}